// SpectralClusteringLoss_7507602833892
// MI455X (gfx1250) — compile-verified
//
#include <hip/hip_runtime.h>
#include <hip/hip_bf16.h>

// SpectralClusteringLoss, algebraically reduced:
//   loss = sum(G) - trace(G),  G = M M^T,  M[c] = sum of feature rows in cluster c.
// Pass 1: bandwidth-bound segmented sum (8 MB features -> 64x256 M), deterministic
//         (thread t exclusively owns LDS column t; fixed-order cross-block reduce).
// Pass 2: 64x64 Gram via v_wmma_f32_16x16x4_f32 (16 waves, one 16x16 tile each),
//         then sum-minus-diagonal reduction to the scalar loss.

typedef __attribute__((ext_vector_type(2))) float v2f;
typedef __attribute__((ext_vector_type(8))) float v8f;

#define NFEAT   256
#define NCLUST  64
#define NROWS   8192
#define MSTRIDE 258   // padded LDS row stride (floats) to spread banks

// ---------------- Pass 1: per-block cluster sums -> partials[b][64][256] ----
__global__ void clusterSumKernel(const float* __restrict__ feats,
                                 const int* __restrict__ assign,
                                 float* __restrict__ partials,
                                 int chunk) {
    __shared__ float sh[NCLUST * NFEAT];        // 64 KB
    const int t = threadIdx.x;                  // t == column, 256 threads
    const int b = blockIdx.x;

    for (int c = 0; c < NCLUST; ++c) sh[c * NFEAT + t] = 0.0f;
    __syncthreads();

    const int row0 = b * chunk;
    for (int i = 0; i < chunk; ++i) {
        const int r = row0 + i;
        if (r >= NROWS) break;
        const int c = assign[r];                // uniform across the wave
        sh[c * NFEAT + t] += feats[r * NFEAT + t];  // thread-private column: no race
    }
    __syncthreads();

    float* dst = partials + b * (NCLUST * NFEAT);
    for (int c = 0; c < NCLUST; ++c) dst[c * NFEAT + t] = sh[c * NFEAT + t];
}

// ---------------- Pass 2: reduce partials, Gram via WMMA, final scalar ------
__global__ void gramLossKernel(const float* __restrict__ partials,
                               float* __restrict__ out,
                               int nb) {
    __shared__ float sh[NCLUST * MSTRIDE];      // ~64.5 KB padded M
    const int tid = threadIdx.x;                // 512 threads = 16 waves

    // Fixed-order reduction over blocks -> deterministic M in LDS.
    for (int j = tid; j < NCLUST * NFEAT; j += 512) {
        float s = 0.0f;
        for (int b = 0; b < nb; ++b) s += partials[b * (NCLUST * NFEAT) + j];
        sh[(j >> 8) * MSTRIDE + (j & 255)] = s;
    }
    __syncthreads();

    // One 16x16 tile of G = M M^T per wave (4x4 tile grid covers 64x64).
    const int wave = tid >> 5;
    const int lane = tid & 31;
    const int ci = wave >> 2;
    const int cj = wave & 3;
    const int n  = lane & 15;     // row-within-tile for A frag / col for B frag
    const int hi = lane >> 4;     // selects K pair {0,1} vs {2,3}

    const float* rowA = &sh[(ci * 16 + n) * MSTRIDE];
    const float* rowB = &sh[(cj * 16 + n) * MSTRIDE];

    v8f acc = {0.f, 0.f, 0.f, 0.f, 0.f, 0.f, 0.f, 0.f};
    for (int k = 0; k < NFEAT; k += 4) {
        // A (16x4 f32) and B (4x16 f32, = rows of M[cj-block]) share the same
        // per-lane layout for M*M^T: float2 at [base + (lane&15)][k + 2*(lane>>4)]
        v2f a = *(const v2f*)(rowA + k + 2 * hi);
        v2f b = *(const v2f*)(rowB + k + 2 * hi);
        acc = __builtin_amdgcn_wmma_f32_16x16x4_f32(
            /*neg_a=*/false, a, /*neg_b=*/false, b,
            /*c_mod=*/(short)0, acc, /*reuse_a=*/false, /*reuse_b=*/false);
    }

    // Lane holds G entries (M = v + 8*hi, N = n). Want sum(G) - trace(G):
    // diagonal entries contribute net zero.
    float p = acc[0] + acc[1] + acc[2] + acc[3] + acc[4] + acc[5] + acc[6] + acc[7];
    if (ci == cj) {
        const int dv = n - 8 * hi;
        if (dv >= 0 && dv < 8) p -= acc[dv];
    }

    // Wave32 reduction.
    for (int m = 16; m >= 1; m >>= 1) p += __shfl_xor(p, m, 32);

    __syncthreads();                 // all LDS reads of M are done; safe to reuse
    if (lane == 0) sh[wave] = p;
    __syncthreads();
    if (tid == 0) {
        float s = 0.0f;
        for (int w = 0; w < 16; ++w) s += sh[w];   // fixed order -> deterministic
        out[0] = s;
    }
}

extern "C" void kernel_launch(void* const* d_in, const int* in_sizes, int n_in,
                              void* d_out, int out_size, void* d_ws, size_t ws_size,
                              hipStream_t stream) {
    const float* feats  = (const float*)d_in[0];   // 8192 x 256 fp32
    const int*   assign = (const int*)d_in[1];     // 8192 cluster ids
    float* out      = (float*)d_out;               // scalar fp32 loss
    float* partials = (float*)d_ws;                // nb * 64 * 256 fp32

    // Adaptive block count so partials fit in whatever scratch we were given.
    int nb = 64;                                   // 4 MB of partials at nb=64
    while (nb > 1 && ws_size < (size_t)nb * NCLUST * NFEAT * sizeof(float)) nb >>= 1;
    const int chunk = NROWS / nb;

    clusterSumKernel<<<nb, 256, 0, stream>>>(feats, assign, partials, chunk);
    gramLossKernel<<<1, 512, 0, stream>>>(partials, out, nb);
}